// HGSocialRec_74861279969938
// MI455X (gfx1250) — compile-verified
//
#include <hip/hip_runtime.h>
#include <hip/hip_bf16.h>

#define DD 128  // embedding size

typedef float v2f __attribute__((ext_vector_type(2)));
typedef float v8f __attribute__((ext_vector_type(8)));

// ---------------------------------------------------------------------------
// C[M,128] = act(A[M,128] @ W[128,128] + bias)     act: 0=none 1=tanh 2=relu
// One block = 16 output rows x 128 cols; 8 waves, each wave one 16x16 tile
// via V_WMMA_F32_16X16X4_F32 (K-loop of 32). W and the A-strip live in LDS.
// Requires M % 16 == 0 (N=100000, 2N=200000 both divisible).
// ---------------------------------------------------------------------------
__global__ __launch_bounds__(256) void gemm128_wmma_kernel(
    const float* __restrict__ A, const float* __restrict__ W,
    const float* __restrict__ bias, float* __restrict__ C,
    int M, int act)
{
    __shared__ float Ws[DD * DD];   // 64 KB
    __shared__ float As[16 * DD];   // 8 KB
    const int tid = threadIdx.x;
    const int rowBase = blockIdx.x * 16;

    // Stage W (16384 floats) into LDS, coalesced float4.
    {
        const float4* Wg = (const float4*)W;
        float4*       Wl = (float4*)Ws;
        #pragma unroll
        for (int i = 0; i < 16; ++i)
            Wl[tid + 256 * i] = Wg[tid + 256 * i];
    }
    // Stage the 16x128 A strip (2048 floats).
    {
        const float4* Ag = (const float4*)(A + (size_t)rowBase * DD);
        float4*       Al = (float4*)As;
        Al[tid]       = Ag[tid];
        Al[tid + 256] = Ag[tid + 256];
    }
    __syncthreads();

    const int wave = tid >> 5;
    const int lane = tid & 31;
    const int half = lane >> 4;   // 0: K pair {k,k+1}; 1: K pair {k+2,k+3}
    const int l16  = lane & 15;
    const int colBase = wave * 16;

    v8f c = {};
    #pragma unroll
    for (int k = 0; k < DD; k += 4) {
        // A fragment 16x4: lane row = l16, VGPR0/1 = K {k+2*half, k+2*half+1}
        v2f a = *(const v2f*)&As[l16 * DD + k + 2 * half];
        // B fragment 4x16: lane col = colBase+l16, rows K {k+2*half, +1}
        v2f b;
        b.x = Ws[(k + 2 * half)     * DD + colBase + l16];
        b.y = Ws[(k + 2 * half + 1) * DD + colBase + l16];
        c = __builtin_amdgcn_wmma_f32_16x16x4_f32(
                /*neg_a=*/false, a, /*neg_b=*/false, b,
                /*c_mod=*/(short)0, c, /*reuse_a=*/false, /*reuse_b=*/false);
    }

    const float bv = bias ? bias[colBase + l16] : 0.0f;
    #pragma unroll
    for (int i = 0; i < 8; ++i) {
        // C layout: VGPR i -> M = i (lanes 0-15) / 8+i (lanes 16-31), N = l16
        const int row = rowBase + i + 8 * half;
        float v = c[i] + bv;
        if (act == 1)      v = tanhf(v);
        else if (act == 2) v = fmaxf(v, 0.0f);
        C[(size_t)row * DD + colBase + l16] = v;
    }
}

// ---------------------------------------------------------------------------
// Gated fusion: score_v = softmax_v( dot(T[v,n,:], w2) + b2 );
// fused[n,:] = x0*s0 + x1*s1.  One wave32 per node, 4 floats per lane.
// ---------------------------------------------------------------------------
__global__ __launch_bounds__(256) void fusion_kernel(
    const float* __restrict__ x, const float* __restrict__ T,
    const float* __restrict__ w2, const float* __restrict__ b2,
    float* __restrict__ fused, int N_)
{
    const int wave = threadIdx.x >> 5, lane = threadIdx.x & 31;
    const int n = blockIdx.x * 8 + wave;
    if (n >= N_) return;

    const float4 a0 = ((const float4*)(T + (size_t)n * DD))[lane];
    const float4 a1 = ((const float4*)(T + (size_t)(N_ + n) * DD))[lane];
    const float4 wv = ((const float4*)w2)[lane];

    float p0 = a0.x * wv.x + a0.y * wv.y + a0.z * wv.z + a0.w * wv.w;
    float p1 = a1.x * wv.x + a1.y * wv.y + a1.z * wv.z + a1.w * wv.w;
    #pragma unroll
    for (int m = 16; m; m >>= 1) {
        p0 += __shfl_xor(p0, m, 32);
        p1 += __shfl_xor(p1, m, 32);
    }
    const float b = b2[0];
    const float s0 = p0 + b, s1 = p1 + b;
    const float mx = fmaxf(s0, s1);
    const float e0 = __expf(s0 - mx), e1 = __expf(s1 - mx);
    const float inv = 1.0f / (e0 + e1);
    const float g0 = e0 * inv, g1 = e1 * inv;

    const float4 v0 = ((const float4*)(x + (size_t)n * DD))[lane];
    const float4 v1 = ((const float4*)(x + (size_t)(N_ + n) * DD))[lane];
    float4 o;
    o.x = v0.x * g0 + v1.x * g1;
    o.y = v0.y * g0 + v1.y * g1;
    o.z = v0.z * g0 + v1.z * g1;
    o.w = v0.w * g0 + v1.w * g1;
    ((float4*)(fused + (size_t)n * DD))[lane] = o;
}

// ---------------------------------------------------------------------------
// SpMM scatter: Y[rows[e],:] += vals[e] * X[cols[e],:]. One wave32 per edge,
// 4 floats per lane, L2 fp32 atomics (relaxed, agent scope).
// ---------------------------------------------------------------------------
__global__ __launch_bounds__(256) void spmm_kernel(
    const int* __restrict__ rows, const int* __restrict__ cols,
    const float* __restrict__ vals, const float* __restrict__ X,
    float* __restrict__ Y, int E_)
{
    const int wave = threadIdx.x >> 5, lane = threadIdx.x & 31;
    const int e = blockIdx.x * 8 + wave;
    if (e >= E_) return;
    const int   r = rows[e];
    const int   c = cols[e];
    const float v = vals[e];
    const float4 xv = ((const float4*)(X + (size_t)c * DD))[lane];
    float* y = Y + (size_t)r * DD + lane * 4;
    __hip_atomic_fetch_add(y + 0, v * xv.x, __ATOMIC_RELAXED, __HIP_MEMORY_SCOPE_AGENT);
    __hip_atomic_fetch_add(y + 1, v * xv.y, __ATOMIC_RELAXED, __HIP_MEMORY_SCOPE_AGENT);
    __hip_atomic_fetch_add(y + 2, v * xv.z, __ATOMIC_RELAXED, __HIP_MEMORY_SCOPE_AGENT);
    __hip_atomic_fetch_add(y + 3, v * xv.w, __ATOMIC_RELAXED, __HIP_MEMORY_SCOPE_AGENT);
}

// ---------------------------------------------------------------------------
// Elementwise helpers (grid-stride).
// ---------------------------------------------------------------------------
__global__ void zero_kernel(float* __restrict__ p, int n) {
    for (int i = blockIdx.x * blockDim.x + threadIdx.x; i < n; i += gridDim.x * blockDim.x)
        p[i] = 0.0f;
}
__global__ void relu_kernel(float* __restrict__ p, int n) {
    for (int i = blockIdx.x * blockDim.x + threadIdx.x; i < n; i += gridDim.x * blockDim.x)
        p[i] = fmaxf(p[i], 0.0f);
}
__global__ void copy_kernel(float* __restrict__ d, const float* __restrict__ s, int n) {
    for (int i = blockIdx.x * blockDim.x + threadIdx.x; i < n; i += gridDim.x * blockDim.x)
        d[i] = s[i];
}
__global__ void add_kernel(float* __restrict__ d, const float* __restrict__ s, int n) {
    for (int i = blockIdx.x * blockDim.x + threadIdx.x; i < n; i += gridDim.x * blockDim.x)
        d[i] += s[i];
}

extern "C" void kernel_launch(void* const* d_in, const int* in_sizes, int n_in,
                              void* d_out, int out_size, void* d_ws, size_t ws_size,
                              hipStream_t stream) {
    const float* x    = (const float*)d_in[0];
    const float* fw1  = (const float*)d_in[1];
    const float* fb1  = (const float*)d_in[2];
    const float* fw2  = (const float*)d_in[3];
    const float* fb2  = (const float*)d_in[4];
    const float* hg_w = (const float*)d_in[5];   // [3,128,128]
    const float* lg_w = (const float*)d_in[6];   // [2,128,128]
    const float* lg_b = (const float*)d_in[7];   // [2,128]
    const float* c1v  = (const float*)d_in[8];
    const float* c2v  = (const float*)d_in[9];
    const float* lgv  = (const float*)d_in[10];
    const int*   c1r  = (const int*)d_in[11];
    const int*   c1c  = (const int*)d_in[12];
    const int*   c2r  = (const int*)d_in[13];
    const int*   c2c  = (const int*)d_in[14];
    const int*   lgr  = (const int*)d_in[15];
    const int*   lgc  = (const int*)d_in[16];

    const int N_ = in_sizes[0] / (2 * DD);   // 100000
    const int E_ = in_sizes[8];              // 3200000
    const int ND = N_ * DD;                  // 12.8M floats
    const int WW = DD * DD;                  // one 128x128 weight matrix

    float* out = (float*)d_out;
    float* OH  = out;                        // final h  -> out[0   .. ND)
    float* OA  = out + (size_t)ND;           // final acc-> out[ND .. 2ND)
    float* W0  = (float*)d_ws;               // ws ping buffer  [ND floats]
    float* W1  = W0 + (size_t)ND;            // ws pong buffer  [ND floats]

    const dim3 B(256);
    const int ewBlocks   = 2048;             // grid-stride elementwise
    const int spmmBlocks = (E_ + 7) / 8;     // one wave per edge
    const int fusBlocks  = (N_ + 7) / 8;

    // 1. T = tanh(x_flat[2N,128] @ fw1 + fb1)  -> use d_out (2N*128) as scratch
    gemm128_wmma_kernel<<<(2 * N_) / 16, B, 0, stream>>>(x, fw1, fb1, out, 2 * N_, 1);
    // 2. fused = softmax-gated sum over views  -> W0
    fusion_kernel<<<fusBlocks, B, 0, stream>>>(x, out, fw2, fb2, W0, N_);

    // --- HypergraphConv ---
    // 3. W1 = fused @ hg_w[0]
    gemm128_wmma_kernel<<<N_ / 16, B, 0, stream>>>(W0, hg_w, nullptr, W1, N_, 0);
    // 4. W0 = relu(spmm1(W1))
    zero_kernel<<<ewBlocks, B, 0, stream>>>(W0, ND);
    spmm_kernel<<<spmmBlocks, B, 0, stream>>>(c1r, c1c, c1v, W1, W0, E_);
    relu_kernel<<<ewBlocks, B, 0, stream>>>(W0, ND);
    // 5. W1 = W0 @ hg_w[1]
    gemm128_wmma_kernel<<<N_ / 16, B, 0, stream>>>(W0, hg_w + WW, nullptr, W1, N_, 0);
    // 6. W0 = relu(spmm1(W1))
    zero_kernel<<<ewBlocks, B, 0, stream>>>(W0, ND);
    spmm_kernel<<<spmmBlocks, B, 0, stream>>>(c1r, c1c, c1v, W1, W0, E_);
    relu_kernel<<<ewBlocks, B, 0, stream>>>(W0, ND);
    // 7. W1 = W0 @ hg_w[2]
    gemm128_wmma_kernel<<<N_ / 16, B, 0, stream>>>(W0, hg_w + 2 * WW, nullptr, W1, N_, 0);
    // 8. OH = relu(spmm1(W1))   -> final h, written straight into out
    zero_kernel<<<ewBlocks, B, 0, stream>>>(OH, ND);
    spmm_kernel<<<spmmBlocks, B, 0, stream>>>(c1r, c1c, c1v, W1, OH, E_);
    relu_kernel<<<ewBlocks, B, 0, stream>>>(OH, ND);
    // 9. W1 = y = relu(spmm2(OH))
    zero_kernel<<<ewBlocks, B, 0, stream>>>(W1, ND);
    spmm_kernel<<<spmmBlocks, B, 0, stream>>>(c2r, c2c, c2v, OH, W1, E_);
    relu_kernel<<<ewBlocks, B, 0, stream>>>(W1, ND);
    // 10. acc = y
    copy_kernel<<<ewBlocks, B, 0, stream>>>(OA, W1, ND);

    // --- LineConv: 2 iterations of cur = spmm_lg(cur @ lg_w[i] + lg_b[i]); acc += cur
    for (int i = 0; i < 2; ++i) {
        gemm128_wmma_kernel<<<N_ / 16, B, 0, stream>>>(W1, lg_w + i * WW, lg_b + i * DD, W0, N_, 0);
        zero_kernel<<<ewBlocks, B, 0, stream>>>(W1, ND);
        spmm_kernel<<<spmmBlocks, B, 0, stream>>>(lgr, lgc, lgv, W0, W1, E_);
        add_kernel<<<ewBlocks, B, 0, stream>>>(OA, W1, ND);
    }
    (void)n_in; (void)out_size; (void)ws_size;
}